// PointStretchLoss_1382979470105
// MI455X (gfx1250) — compile-verified
//
#include <hip/hip_runtime.h>
#include <math.h>

typedef __attribute__((ext_vector_type(2))) float v2f;
typedef __attribute__((ext_vector_type(8))) float v8f;

#define NPTS 4096
#define KNN  16      // neighbors kept (nn_size)
#define KSEL 17      // including self
#define ROWS_PER_WG 32
#define EPSV 1e-10f

__global__ void zero_out_kernel(float* out) { out[0] = 0.0f; }

// One wave32 per 32 rows of one batch.  Gram matrix tiles via
// V_WMMA_F32_16X16X4_F32 (A = 16x4 f32, D=3 padded to 4 with zeros).
__global__ __launch_bounds__(32)
void point_stretch_kernel(const float* __restrict__ ref,
                          const float* __restrict__ pred,
                          float* __restrict__ out,
                          int nBatch)
{
    const int lane = threadIdx.x;     // 0..31
    const int l    = lane & 15;
    const int half = lane >> 4;       // 0: lanes 0-15, 1: lanes 16-31

    const int wavesPerBatch = NPTS / ROWS_PER_WG;   // 128
    const int b  = blockIdx.x / wavesPerBatch;
    const int i0 = (blockIdx.x % wavesPerBatch) * ROWS_PER_WG;
    if (b >= nBatch) return;          // uniform per block

    const float* refB  = ref  + (size_t)b * NPTS * 3;
    const float* predB = pred + (size_t)b * NPTS * 3;

    __shared__ float sqrow[ROWS_PER_WG];
    __shared__ float d2s[ROWS_PER_WG * 16];

    // ---- A fragments (two 16-row tiles) ----
    // ISA 32-bit A 16x4 layout: lanes 0-15: VGPR0=K0, VGPR1=K1 for M=lane;
    // lanes 16-31: VGPR0=K2, VGPR1=K3 for M=lane-16.  K3 = 0 (D=3 pad).
    const float* r0 = refB + (size_t)(i0 + l) * 3;
    const float* r1 = refB + (size_t)(i0 + 16 + l) * 3;
    float a0x = r0[0], a0y = r0[1], a0z = r0[2];
    float a1x = r1[0], a1y = r1[1], a1z = r1[2];
    v2f a0, a1;
    a0.x = half ? a0z : a0x;  a0.y = half ? 0.0f : a0y;
    a1.x = half ? a1z : a1x;  a1.y = half ? 0.0f : a1y;

    // squared norm of this lane's own row (row i0+lane), shared to all lanes
    const float* rown = refB + (size_t)(i0 + lane) * 3;
    float rx = rown[0], ry = rown[1], rz = rown[2];
    sqrow[lane] = rx * rx + ry * ry + rz * rz;
    __syncthreads();

    float sqA0[8], sqA1[8];
    #pragma unroll
    for (int v = 0; v < 8; ++v) {
        sqA0[v] = sqrow[v + 8 * half];        // row m = v + 8*half of tile 0
        sqA1[v] = sqrow[16 + v + 8 * half];   // tile 1
    }

    // ---- per-lane sorted top-KSEL list for row (i0 + lane) ----
    float dl[KSEL]; int il[KSEL];
    #pragma unroll
    for (int s = 0; s < KSEL; ++s) { dl[s] = 3.0e38f; il[s] = 0; }

    for (int j0 = 0; j0 < NPTS; j0 += 16) {
        // B fragment: B[k][n] = refB[j0+n][k]; lanes 0-15 hold K0/K1 rows,
        // lanes 16-31 hold K2/K3 rows (mirrors A layout).
        const float* cp = refB + (size_t)(j0 + l) * 3;
        float bx = cp[0], by = cp[1], bz = cp[2];
        v2f bf;  bf.x = half ? bz : bx;  bf.y = half ? 0.0f : by;
        float sqcol = bx * bx + by * by + bz * bz;   // ||col (j0+l)||^2

        v8f acc0 = {};
        v8f acc1 = {};
        acc0 = __builtin_amdgcn_wmma_f32_16x16x4_f32(false, a0, false, bf,
                                                     (short)0, acc0, false, false);
        acc1 = __builtin_amdgcn_wmma_f32_16x16x4_f32(false, a1, false, bf,
                                                     (short)0, acc1, false, false);

        // C layout: lane holds N = l, M = v + 8*half for VGPR v.
        // d2(m, j0+l) = ||r_m||^2 + ||c||^2 - 2*dot
        #pragma unroll
        for (int v = 0; v < 8; ++v) {
            int m0 = v + 8 * half;
            d2s[m0 * 16 + l]        = sqA0[v] + sqcol - 2.0f * acc0[v];
            d2s[(16 + m0) * 16 + l] = sqA1[v] + sqcol - 2.0f * acc1[v];
        }
        __syncthreads();

        // selection: lane owns row (i0+lane); scan this tile's 16 candidates
        #pragma unroll
        for (int c = 0; c < 16; ++c) {
            float cd = d2s[lane * 16 + c];
            int   ci = j0 + c;
            if (cd < dl[KSEL - 1]) {          // beats current worst -> insert
                bool active = true;
                #pragma unroll
                for (int s = KSEL - 1; s > 0; --s) {
                    bool  sh = active && (cd < dl[s - 1]);
                    float nd = sh ? dl[s - 1] : cd;
                    int   ni = sh ? il[s - 1] : ci;
                    if (active) { dl[s] = nd; il[s] = ni; }
                    active = sh;
                }
                if (active) { dl[0] = cd; il[0] = ci; }
            }
        }
        __syncthreads();   // protect d2s WAR for next tile
    }

    // ---- epilogue: drop self (slot 0), compute stretch over 16 neighbors ----
    const float* po = predB + (size_t)(i0 + lane) * 3;
    float px = po[0], py = po[1], pz = po[2];

    float acc = 0.0f;
    #pragma unroll
    for (int k = 1; k < KSEL; ++k) {
        int   j    = il[k];
        float dref = sqrtf(fmaxf(dl[k], 0.0f));
        const float* pj = predB + (size_t)j * 3;
        float dx = pj[0] - px, dy = pj[1] - py, dz = pj[2] - pz;
        float dist = sqrtf(dx * dx + dy * dy + dz * dz);
        acc += fmaxf(dist / (dref + EPSV) - 1.0f, 0.0f);
    }

    // wave32 reduction
    #pragma unroll
    for (int off = 16; off > 0; off >>= 1)
        acc += __shfl_down(acc, off, 32);

    if (lane == 0) {
        float scale = 1.0f / ((float)nBatch * (float)NPTS * (float)KNN);
        atomicAdd(out, acc * scale);
    }
}

extern "C" void kernel_launch(void* const* d_in, const int* in_sizes, int n_in,
                              void* d_out, int out_size, void* d_ws, size_t ws_size,
                              hipStream_t stream) {
    (void)n_in; (void)out_size; (void)d_ws; (void)ws_size;
    const float* ref  = (const float*)d_in[0];   // points_ref (B,N,3) f32
    const float* pred = (const float*)d_in[1];   // points     (B,N,3) f32
    float* out = (float*)d_out;                  // scalar f32

    int B = in_sizes[0] / (NPTS * 3);
    if (B < 1) B = 1;

    zero_out_kernel<<<1, 1, 0, stream>>>(out);
    int blocks = B * (NPTS / ROWS_PER_WG);       // 1024 for B=8
    point_stretch_kernel<<<blocks, ROWS_PER_WG, 0, stream>>>(ref, pred, out, B);
}